// GATLayer_61323543052451
// MI455X (gfx1250) — compile-verified
//
#include <hip/hip_runtime.h>
#include <hip/hip_bf16.h>

typedef float v2f __attribute__((ext_vector_type(2)));
typedef float v8f __attribute__((ext_vector_type(8)));

static constexpr int Bn = 8;
static constexpr int Nn = 2048;
static constexpr int Dd = 1024;
static constexpr int Hh = 512;

__device__ __forceinline__ v8f wmma_f32(v2f a, v2f b, v8f c) {
  // V_WMMA_F32_16X16X4_F32 : D = A(16x4) * B(4x16) + C, fp32 throughout
  return __builtin_amdgcn_wmma_f32_16x16x4_f32(false, a, false, b, (short)0, c,
                                               false, false);
}

// Enforce the schedule [5 x VMEM-read][4 x WMMA] per pipeline step so the
// scheduler cannot re-serialize load->wait(0)->wmma and the allocator must
// keep the staged fragments in distinct registers.
#if __has_builtin(__builtin_amdgcn_sched_group_barrier)
#define SCHED_5L4W()                                 \
  __builtin_amdgcn_sched_group_barrier(0x020, 5, 0); \
  __builtin_amdgcn_sched_group_barrier(0x008, 4, 0)
#else
#define SCHED_5L4W()
#endif

// ---------------------------------------------------------------------------
// Unified NT GEMM: C[M,N] = A[M,K] * Bt[N,K]^T, one wave owns a 16x64 tile.
// Depth-2 software pipeline: step k issues the 5 b64 fragment loads for k+8
// and runs the 4 WMMAs of step k on fragments loaded two steps earlier
// (10 loads in flight at every wait point).
// EPI: 0 = none, 1 = +bias, 2 = sigmoid.
// ---------------------------------------------------------------------------
template <int EPI>
__global__ __launch_bounds__(128) void gemm16x64_nt(
    const float* __restrict__ A, int lda, const float* __restrict__ Bt,
    int ldb, float* __restrict__ C, int ldc, const float* __restrict__ bias,
    int K, int colGroups) {
  const int lane = threadIdx.x & 31;
  const int wave = threadIdx.x >> 5;
  const int half = lane >> 4;  // selects K pair within the 16x4 fragment
  const int r = lane & 15;     // row (A) / col (B,C) within tile

  const int tile = blockIdx.x * 4 + wave;
  const int m0 = (tile / colGroups) << 4;
  const int n0 = (tile % colGroups) << 6;

  const float* Arow = A + (size_t)(m0 + r) * lda + 2 * half;
  const float* Brow0 = Bt + (size_t)(n0 + 0 + r) * ldb + 2 * half;
  const float* Brow1 = Bt + (size_t)(n0 + 16 + r) * ldb + 2 * half;
  const float* Brow2 = Bt + (size_t)(n0 + 32 + r) * ldb + 2 * half;
  const float* Brow3 = Bt + (size_t)(n0 + 48 + r) * ldb + 2 * half;

  v8f acc0 = {}, acc1 = {}, acc2 = {}, acc3 = {};

  // stage 0: fragments for step k; stage 1: fragments for step k+4
  v2f a0 = *(const v2f*)(Arow);
  v2f b00 = *(const v2f*)(Brow0);
  v2f b01 = *(const v2f*)(Brow1);
  v2f b02 = *(const v2f*)(Brow2);
  v2f b03 = *(const v2f*)(Brow3);
  v2f a1 = *(const v2f*)(Arow + 4);
  v2f b10 = *(const v2f*)(Brow0 + 4);
  v2f b11 = *(const v2f*)(Brow1 + 4);
  v2f b12 = *(const v2f*)(Brow2 + 4);
  v2f b13 = *(const v2f*)(Brow3 + 4);

#define GEMM_STEP(KNEXT)                                                      \
  do {                                                                        \
    v2f an = *(const v2f*)(Arow + (KNEXT));                                   \
    v2f bn0 = *(const v2f*)(Brow0 + (KNEXT));                                 \
    v2f bn1 = *(const v2f*)(Brow1 + (KNEXT));                                 \
    v2f bn2 = *(const v2f*)(Brow2 + (KNEXT));                                 \
    v2f bn3 = *(const v2f*)(Brow3 + (KNEXT));                                 \
    acc0 = wmma_f32(a0, b00, acc0);                                           \
    acc1 = wmma_f32(a0, b01, acc1);                                           \
    acc2 = wmma_f32(a0, b02, acc2);                                           \
    acc3 = wmma_f32(a0, b03, acc3);                                           \
    a0 = a1; b00 = b10; b01 = b11; b02 = b12; b03 = b13;                      \
    a1 = an; b10 = bn0; b11 = bn1; b12 = bn2; b13 = bn3;                      \
    SCHED_5L4W();                                                             \
  } while (0)

#define GEMM_TAIL()                                                           \
  do {                                                                        \
    acc0 = wmma_f32(a0, b00, acc0);                                           \
    acc1 = wmma_f32(a0, b01, acc1);                                           \
    acc2 = wmma_f32(a0, b02, acc2);                                           \
    acc3 = wmma_f32(a0, b03, acc3);                                           \
    a0 = a1; b00 = b10; b01 = b11; b02 = b12; b03 = b13;                      \
  } while (0)

  // main chunks: pipelined loads reach kk+68 <= K-60 (strictly in bounds)
  for (int kk = 0; kk + 64 < K; kk += 64) {
#pragma unroll
    for (int ks = 0; ks < 64; ks += 4) GEMM_STEP(kk + ks + 8);
    // prefetch the chunk three ahead (placed after the group pattern so it
    // cannot be claimed into a VMEM-read scheduling group)
    __builtin_prefetch(Arow + kk + 192, 0, 3);
    __builtin_prefetch(Brow0 + kk + 192, 0, 3);
    __builtin_prefetch(Brow2 + kk + 192, 0, 3);
  }
  // drain chunk: 14 pipelined steps (loads up to base K-4), then 2 dry steps
  {
    const int kk = K - 64;
#pragma unroll
    for (int ks = 0; ks < 56; ks += 4) GEMM_STEP(kk + ks + 8);
    GEMM_TAIL();  // step K-8
    GEMM_TAIL();  // step K-4
  }
#undef GEMM_STEP
#undef GEMM_TAIL

  v8f accs[4] = {acc0, acc1, acc2, acc3};
#pragma unroll
  for (int c = 0; c < 4; ++c) {
#pragma unroll
    for (int v = 0; v < 8; ++v) {
      const int row = m0 + v + 8 * half;
      const int col = n0 + c * 16 + r;
      float val = accs[c][v];
      if (EPI == 1) val += bias[col];
      if (EPI == 2) val = 1.0f / (1.0f + __expf(-val));
      C[(size_t)row * ldc + col] = val;
    }
  }
}

// ---------------------------------------------------------------------------
// 32x32 LDS-tiled transpose: out[C x R] = in[R x C]^T, batched via blockIdx.z
// (batch stride R*C identical on both sides). Block (32,8).
// ---------------------------------------------------------------------------
__global__ __launch_bounds__(256) void transpose_kernel(
    const float* __restrict__ in, float* __restrict__ out, int R, int C) {
  __shared__ float t[32][33];
  const size_t base = (size_t)blockIdx.z * R * C;
  const int c0 = blockIdx.x * 32;
  const int r0 = blockIdx.y * 32;
  const int tx = threadIdx.x;
  const int ty = threadIdx.y;
#pragma unroll
  for (int i = 0; i < 32; i += 8)
    t[ty + i][tx] = in[base + (size_t)(r0 + ty + i) * C + c0 + tx];
  __syncthreads();
#pragma unroll
  for (int i = 0; i < 32; i += 8)
    out[base + (size_t)(c0 + ty + i) * R + r0 + tx] = t[tx][ty + i];
}

// ---------------------------------------------------------------------------
// LayerNorm (in place over H=512) + fused s_i = h.w1, s_j = h.w2
// One wave32 per row; 16 elements per lane.
// ---------------------------------------------------------------------------
__global__ __launch_bounds__(256) void ln_s_kernel(
    float* __restrict__ h, const float* __restrict__ gamma,
    const float* __restrict__ beta, const float* __restrict__ attn_w,
    float* __restrict__ s_i, float* __restrict__ s_j) {
  const int lane = threadIdx.x & 31;
  const int wave = threadIdx.x >> 5;
  const int row = blockIdx.x * 8 + wave;
  float* hr = h + (size_t)row * Hh;

  float x[16];
  float sum = 0.0f;
#pragma unroll
  for (int i = 0; i < 16; ++i) {
    x[i] = hr[lane + 32 * i];
    sum += x[i];
  }
#pragma unroll
  for (int o = 16; o > 0; o >>= 1) sum += __shfl_xor(sum, o, 32);
  const float mean = sum * (1.0f / 512.0f);

  float vs = 0.0f;
#pragma unroll
  for (int i = 0; i < 16; ++i) {
    const float d = x[i] - mean;
    vs += d * d;
  }
#pragma unroll
  for (int o = 16; o > 0; o >>= 1) vs += __shfl_xor(vs, o, 32);
  const float rstd = rsqrtf(vs * (1.0f / 512.0f) + 1e-5f);

  float si = 0.0f, sj = 0.0f;
#pragma unroll
  for (int i = 0; i < 16; ++i) {
    const int col = lane + 32 * i;
    const float y = (x[i] - mean) * rstd * gamma[col] + beta[col];
    hr[col] = y;
    si += y * attn_w[col];
    sj += y * attn_w[Hh + col];
  }
#pragma unroll
  for (int o = 16; o > 0; o >>= 1) {
    si += __shfl_xor(si, o, 32);
    sj += __shfl_xor(sj, o, 32);
  }
  if (lane == 0) {
    s_i[row] = si;
    s_j[row] = sj;
  }
}

// ---------------------------------------------------------------------------
// Per-row 0.7-quantile of a 2048-wide row via in-LDS bitonic sort.
// jnp.quantile linear method: pos = 0.7*2047 = 1432.9 -> lerp(s[1432],s[1433],0.9)
// ---------------------------------------------------------------------------
__global__ __launch_bounds__(1024) void quantile_kernel(
    const float* __restrict__ Ab, float* __restrict__ delta) {
  __shared__ float s[2048];
  const int t = threadIdx.x;
  const float* row = Ab + (size_t)blockIdx.x * 2048;
  s[t] = row[t];
  s[t + 1024] = row[t + 1024];
  __syncthreads();

  for (int k = 2; k <= 2048; k <<= 1) {
    for (int j = k >> 1; j > 0; j >>= 1) {
      const int i = 2 * t - (t & (j - 1));
      const int ixj = i | j;
      const bool up = ((i & k) == 0);
      const float a = s[i];
      const float b = s[ixj];
      const bool sw = up ? (a > b) : (a < b);
      if (sw) {
        s[i] = b;
        s[ixj] = a;
      }
      __syncthreads();
    }
  }
  if (t == 0) {
    const float lo = s[1432];
    const float hi = s[1433];
    delta[blockIdx.x] = lo + 0.9f * (hi - lo);
  }
}

// ---------------------------------------------------------------------------
// Masked softmax of rank-1 leaky-relu logits; writes normalized alpha
// in place over A (invalid entries -> 0). One wave32 per row.
// ---------------------------------------------------------------------------
__global__ __launch_bounds__(256) void attn_alpha_kernel(
    float* __restrict__ Ab, const float* __restrict__ s_i,
    const float* __restrict__ s_j, const float* __restrict__ delta,
    const float* __restrict__ attn_b) {
  const int lane = threadIdx.x & 31;
  const int wave = threadIdx.x >> 5;
  const int row = blockIdx.x * 8 + wave;
  float* Ar = Ab + (size_t)row * 2048;
  const float si = s_i[row];
  const float d = delta[row];
  const float ab = attn_b[0];

  float mx = -3.402823e38f;
  for (int m = lane; m < 2048; m += 32) {
    const float a = Ar[m];
    const bool valid = (a > d) || (m == row);
    float e = si + s_j[m] + ab;
    e = (e > 0.0f) ? e : 0.01f * e;
    if (valid && e > mx) mx = e;
  }
#pragma unroll
  for (int o = 16; o > 0; o >>= 1) mx = fmaxf(mx, __shfl_xor(mx, o, 32));

  float sum = 0.0f;
  for (int m = lane; m < 2048; m += 32) {
    const float a = Ar[m];
    const bool valid = (a > d) || (m == row);
    float e = si + s_j[m] + ab;
    e = (e > 0.0f) ? e : 0.01f * e;
    sum += valid ? __expf(e - mx) : 0.0f;
  }
#pragma unroll
  for (int o = 16; o > 0; o >>= 1) sum += __shfl_xor(sum, o, 32);
  const float inv = 1.0f / sum;

  for (int m = lane; m < 2048; m += 32) {
    const float a = Ar[m];
    const bool valid = (a > d) || (m == row);
    float e = si + s_j[m] + ab;
    e = (e > 0.0f) ? e : 0.01f * e;
    Ar[m] = valid ? __expf(e - mx) * inv : 0.0f;
  }
}

// ---------------------------------------------------------------------------
extern "C" void kernel_launch(void* const* d_in, const int* in_sizes, int n_in,
                              void* d_out, int out_size, void* d_ws,
                              size_t ws_size, hipStream_t stream) {
  (void)in_sizes; (void)n_in; (void)out_size; (void)ws_size;
  const float* x      = (const float*)d_in[0];  // (B,N,D)
  const float* fc_w   = (const float*)d_in[1];  // (D,H)
  const float* fc_b   = (const float*)d_in[2];  // (H,)
  const float* ln_g   = (const float*)d_in[3];  // (H,)
  const float* ln_b   = (const float*)d_in[4];  // (H,)
  const float* adj_w  = (const float*)d_in[5];  // (H,H)
  const float* attn_w = (const float*)d_in[6];  // (2H,)
  const float* attn_b = (const float*)d_in[7];  // scalar
  float* out = (float*)d_out;                   // (B,N,H)

  char* ws = (char*)d_ws;
  size_t off = 0;
  float* h      = (float*)(ws + off); off += (size_t)Bn * Nn * Hh * 4;  // 33.5M
  float* hT     = (float*)(ws + off); off += (size_t)Bn * Hh * Nn * 4;  // 33.5M
  float* Abuf   = (float*)(ws + off); off += (size_t)Nn * Nn * 4;       // 16.8M
  float* fc_wT  = (float*)(ws + off); off += (size_t)Dd * Hh * 4;       //  2M
  float* adj_wT = (float*)(ws + off); off += (size_t)Hh * Hh * 4;       //  1M
  float* s_i    = (float*)(ws + off); off += (size_t)Bn * Nn * 4;
  float* s_j    = (float*)(ws + off); off += (size_t)Bn * Nn * 4;
  float* delta  = (float*)(ws + off); off += (size_t)Bn * Nn * 4;

  // 0) transpose the weight panels once
  transpose_kernel<<<dim3(Hh / 32, Dd / 32, 1), dim3(32, 8), 0, stream>>>(
      fc_w, fc_wT, Dd, Hh);  // (D,H) -> (H,D)
  transpose_kernel<<<dim3(Hh / 32, Hh / 32, 1), dim3(32, 8), 0, stream>>>(
      adj_w, adj_wT, Hh, Hh);  // (H,H) -> (H,H)^T

  // 1) h_raw = X @ fc_w + fc_b            (NT: Bt = fc_wT)
  {
    const int blocks = ((Bn * Nn / 16) * (Hh / 64)) / 4;  // 2048
    gemm16x64_nt<1><<<blocks, 128, 0, stream>>>(x, Dd, fc_wT, Dd, h, Hh, fc_b,
                                                Dd, Hh / 64);
  }
  // 2) LayerNorm in place + s_i, s_j
  ln_s_kernel<<<Bn * Nn / 8, 256, 0, stream>>>(h, ln_g, ln_b, attn_w, s_i, s_j);

  // 2b) hT[b] = h[b]^T  (per batch (N,H) -> (H,N)); needed by alpha @ h
  transpose_kernel<<<dim3(Hh / 32, Nn / 32, Bn), dim3(32, 8), 0, stream>>>(
      h, hT, Nn, Hh);

  // 3) g = h @ adj_w  (parked in d_out; consumed per batch before overwrite)
  {
    const int blocks = ((Bn * Nn / 16) * (Hh / 64)) / 4;  // 2048
    gemm16x64_nt<0><<<blocks, 128, 0, stream>>>(h, Hh, adj_wT, Hh, out, Hh,
                                                nullptr, Hh, Hh / 64);
  }

  // 4) per-batch pipeline
  for (int b = 0; b < Bn; ++b) {
    const float* gb  = out + (size_t)b * Nn * Hh;
    const float* hb  = h + (size_t)b * Nn * Hh;
    const float* hTb = hT + (size_t)b * Hh * Nn;

    // A = sigmoid(g_b @ h_b^T)   (naturally NT: Bt = h_b)
    {
      const int blocks = ((Nn / 16) * (Nn / 64)) / 4;  // 1024
      gemm16x64_nt<2><<<blocks, 128, 0, stream>>>(gb, Hh, hb, Hh, Abuf, Nn,
                                                  nullptr, Hh, Nn / 64);
    }
    // per-row 0.7-quantile threshold
    quantile_kernel<<<Nn, 1024, 0, stream>>>(Abuf, delta + (size_t)b * Nn);
    // masked softmax -> alpha written in place over A
    attn_alpha_kernel<<<Nn / 8, 256, 0, stream>>>(
        Abuf, s_i + (size_t)b * Nn, s_j + (size_t)b * Nn,
        delta + (size_t)b * Nn, attn_b);
    // out_b = alpha @ h_b         (NT: Bt = hT_b)
    {
      const int blocks = ((Nn / 16) * (Hh / 64)) / 4;  // 256
      gemm16x64_nt<0><<<blocks, 128, 0, stream>>>(
          Abuf, Nn, hTb, Nn, out + (size_t)b * Nn * Hh, Hh, nullptr, Nn,
          Hh / 64);
    }
  }
}